// AUVFossen_50096498541360
// MI455X (gfx1250) — compile-verified
//
#include <hip/hip_runtime.h>
#include <math.h>

#define TPB   256
#define DT_C  0.1f
#define GRAV  9.81f
#define DENS  1028.0f

typedef __attribute__((ext_vector_type(2))) float v2f;
typedef __attribute__((ext_vector_type(8))) float v8f;

// ---------------------------------------------------------------------------
// Prep kernel (1 wave): invert the shared 6x6 mTot.
// Robust scalar Gauss-Jordan, then 2 Newton-Schulz refinement steps
// X <- X(2I - MX) computed with V_WMMA_F32_16X16X4_F32 (6x6 padded to 16x16,
// K contracted in two 4-wide chunks). Thread 0 keeps whichever of
// {GJ result, WMMA-refined result} has the smaller ||I - MX||_inf, so an
// incorrect operand-layout assumption can never corrupt the output.
// ---------------------------------------------------------------------------
__global__ void prep_inverse(const float* __restrict__ mTot, float* __restrict__ ws)
{
    __shared__ float Ms[36], X0[36], Xc[36], Ts[36];
    const int lane = threadIdx.x;
    float r_gj = 0.0f;

    if (lane == 0) {
        for (int i = 0; i < 36; ++i) Ms[i] = mTot[i];
        // Gauss-Jordan with partial pivoting on [A | I]
        float a[6][12];
        for (int i = 0; i < 6; ++i)
            for (int j = 0; j < 6; ++j) {
                a[i][j]     = Ms[i*6 + j];
                a[i][6 + j] = (i == j) ? 1.0f : 0.0f;
            }
        for (int c = 0; c < 6; ++c) {
            int p = c; float mx = fabsf(a[c][c]);
            for (int r = c + 1; r < 6; ++r) {
                float vv = fabsf(a[r][c]);
                if (vv > mx) { mx = vv; p = r; }
            }
            if (p != c)
                for (int j = 0; j < 12; ++j) { float t = a[c][j]; a[c][j] = a[p][j]; a[p][j] = t; }
            float inv = 1.0f / a[c][c];
            for (int j = 0; j < 12; ++j) a[c][j] *= inv;
            for (int r = 0; r < 6; ++r) if (r != c) {
                float f = a[r][c];
                for (int j = 0; j < 12; ++j) a[r][j] = fmaf(-f, a[c][j], a[r][j]);
            }
        }
        for (int i = 0; i < 6; ++i)
            for (int j = 0; j < 6; ++j) {
                X0[i*6 + j] = a[i][6 + j];
                Xc[i*6 + j] = a[i][6 + j];
            }
        // residual of GJ result
        for (int i = 0; i < 6; ++i)
            for (int j = 0; j < 6; ++j) {
                float s = (i == j) ? -1.0f : 0.0f;
                for (int k = 0; k < 6; ++k) s = fmaf(Ms[i*6 + k], X0[k*6 + j], s);
                r_gj = fmaxf(r_gj, fabsf(s));
            }
    }
    __syncthreads();

    const int m  = lane & 15;   // A-row / B-col / C-col index
    const int hi = lane >> 4;   // lane-group select

    for (int it = 0; it < 2; ++it) {
        // T = M @ X  (16x16 padded, K = 0..7 in two chunks of 4)
        v8f acc = {};
#pragma unroll
        for (int kb = 0; kb < 8; kb += 4) {
            v2f A, B;
            int k0 = kb + 2*hi;
            // A f32 16x4: lane&15 = M row; VGPR v holds K = kb + v + 2*(lane>>4)
            A[0] = (m < 6 && k0     < 6) ? Ms[m*6 + k0    ] : 0.0f;
            A[1] = (m < 6 && k0 + 1 < 6) ? Ms[m*6 + k0 + 1] : 0.0f;
            // B f32 4x16: lane&15 = N col; VGPR v holds K = kb + v + 2*(lane>>4)
            B[0] = (m < 6 && k0     < 6) ? Xc[k0*6 + m]       : 0.0f;
            B[1] = (m < 6 && k0 + 1 < 6) ? Xc[(k0 + 1)*6 + m] : 0.0f;
            acc = __builtin_amdgcn_wmma_f32_16x16x4_f32(false, A, false, B,
                                                        (short)0, acc, false, false);
        }
        __syncthreads();
        // R = 2I - T  (C/D layout: VGPR v -> row v + 8*(lane>>4), col lane&15)
#pragma unroll
        for (int v = 0; v < 8; ++v) {
            int mm = v + 8*hi;
            if (mm < 6 && m < 6) Ts[mm*6 + m] = ((mm == m) ? 2.0f : 0.0f) - acc[v];
        }
        __syncthreads();
        // Xnew = X @ R
        v8f acc2 = {};
#pragma unroll
        for (int kb = 0; kb < 8; kb += 4) {
            v2f A, B;
            int k0 = kb + 2*hi;
            A[0] = (m < 6 && k0     < 6) ? Xc[m*6 + k0    ] : 0.0f;
            A[1] = (m < 6 && k0 + 1 < 6) ? Xc[m*6 + k0 + 1] : 0.0f;
            B[0] = (m < 6 && k0     < 6) ? Ts[k0*6 + m]       : 0.0f;
            B[1] = (m < 6 && k0 + 1 < 6) ? Ts[(k0 + 1)*6 + m] : 0.0f;
            acc2 = __builtin_amdgcn_wmma_f32_16x16x4_f32(false, A, false, B,
                                                         (short)0, acc2, false, false);
        }
        __syncthreads();
#pragma unroll
        for (int v = 0; v < 8; ++v) {
            int mm = v + 8*hi;
            if (mm < 6 && m < 6) Xc[mm*6 + m] = acc2[v];
        }
        __syncthreads();
    }

    if (lane == 0) {
        float r_new = 0.0f; bool ok = true;
        for (int i = 0; i < 6; ++i)
            for (int j = 0; j < 6; ++j) {
                float xv = Xc[i*6 + j];
                if (!(xv == xv) || fabsf(xv) > 1e30f) ok = false;
                float s = (i == j) ? -1.0f : 0.0f;
                for (int k = 0; k < 6; ++k) s = fmaf(Ms[i*6 + k], Xc[k*6 + j], s);
                r_new = fmaxf(r_new, fabsf(s));
            }
        const float* src = (ok && r_new <= r_gj) ? Xc : X0;
        for (int i = 0; i < 36; ++i) ws[i] = src[i];
    }
}

// ---------------------------------------------------------------------------
// Fossen state derivative for one row (all parameters from LDS).
// sc: [0]=m*g, [1]=rho*V*g, [2..4]=cog, [5..7]=cob
// ---------------------------------------------------------------------------
__device__ __forceinline__ void fossen_dot(
    const float xl[13], const float ul[6],
    const float* __restrict__ sM, const float* __restrict__ slD,
    const float* __restrict__ slDF, const float* __restrict__ sqD6,
    const float* __restrict__ sMi, const float* __restrict__ sc,
    float xd[13])
{
    const float qx = xl[3], qy = xl[4], qz = xl[5], qw = xl[6];
    const float r00 = 1.0f - 2.0f*(qy*qy + qz*qz);
    const float r01 = 2.0f*(qx*qy - qz*qw);
    const float r02 = 2.0f*(qx*qz + qy*qw);
    const float r10 = 2.0f*(qx*qy + qz*qw);
    const float r11 = 1.0f - 2.0f*(qx*qx + qz*qz);
    const float r12 = 2.0f*(qy*qz - qx*qw);
    const float r20 = 2.0f*(qx*qz - qy*qw);
    const float r21 = 2.0f*(qy*qz + qx*qw);
    const float r22 = 1.0f - 2.0f*(qx*qx + qy*qy);

    float v[6];
#pragma unroll
    for (int i = 0; i < 6; ++i) v[i] = xl[7 + i];
    const float w0 = v[3], w1 = v[4], w2 = v[5];

    // pDot_lin = rot @ v_lin
    xd[0] = fmaf(r00, v[0], fmaf(r01, v[1], r02*v[2]));
    xd[1] = fmaf(r10, v[0], fmaf(r11, v[1], r12*v[2]));
    xd[2] = fmaf(r20, v[0], fmaf(r21, v[1], r22*v[2]));
    // pDot_ang = T(q) @ w
    xd[3] = 0.5f*fmaf(-qx, w0, fmaf(-qy, w1, -qz*w2));
    xd[4] = 0.5f*fmaf( qw, w0, fmaf(-qz, w1,  qy*w2));
    xd[5] = 0.5f*fmaf( qz, w0, fmaf( qw, w1, -qx*w2));
    xd[6] = 0.5f*fmaf(-qy, w0, fmaf( qx, w1,  qw*w2));

    // Dv_i = -(lD_i. v) - v_i (lDF_i. v) - qD_ii |v_i| v_i ;  mv = M v
    float Dv[6], mv[6];
#pragma unroll
    for (int i = 0; i < 6; ++i) {
        float a = 0.0f, b = 0.0f, c = 0.0f;
#pragma unroll
        for (int j = 0; j < 6; ++j) {
            a = fmaf(slD [i*6 + j], v[j], a);
            b = fmaf(slDF[i*6 + j], v[j], b);
            c = fmaf(sM  [i*6 + j], v[j], c);
        }
        Dv[i] = -a - v[i]*b - sqD6[i]*fabsf(v[i])*v[i];
        mv[i] = c;
    }

    // Coriolis: Cv_top = w x mv_top ; Cv_bot = v_lin x mv_top + w x mv_bot
    float Cv[6];
    Cv[0] = fmaf(w1, mv[2], -w2*mv[1]);
    Cv[1] = fmaf(w2, mv[0], -w0*mv[2]);
    Cv[2] = fmaf(w0, mv[1], -w1*mv[0]);
    Cv[3] = fmaf(v[1], mv[2], -v[2]*mv[1]) + fmaf(w1, mv[5], -w2*mv[4]);
    Cv[4] = fmaf(v[2], mv[0], -v[0]*mv[2]) + fmaf(w2, mv[3], -w0*mv[5]);
    Cv[5] = fmaf(v[0], mv[1], -v[1]*mv[0]) + fmaf(w0, mv[4], -w1*mv[3]);

    // Restoring: fbg = rot^T*(0,0,-mg), fbb = rot^T*(0,0,rhoVg) -> rows of rot[2]
    const float fzg = -sc[0];
    const float fzb =  sc[1];
    const float fg0 = r20*fzg, fg1 = r21*fzg, fg2 = r22*fzg;
    const float fb0 = r20*fzb, fb1 = r21*fzb, fb2 = r22*fzb;
    const float cg0 = sc[2], cg1 = sc[3], cg2 = sc[4];
    const float cb0 = sc[5], cb1 = sc[6], cb2 = sc[7];
    float gvec[6];
    gvec[0] = -(fg0 + fb0);
    gvec[1] = -(fg1 + fb1);
    gvec[2] = -(fg2 + fb2);
    gvec[3] = -(fmaf(cg1, fg2, -cg2*fg1) + fmaf(cb1, fb2, -cb2*fb1));
    gvec[4] = -(fmaf(cg2, fg0, -cg0*fg2) + fmaf(cb2, fb0, -cb0*fb2));
    gvec[5] = -(fmaf(cg0, fg1, -cg1*fg0) + fmaf(cb0, fb1, -cb1*fb0));

    float rhs[6];
#pragma unroll
    for (int j = 0; j < 6; ++j) rhs[j] = ul[j] - Cv[j] - Dv[j] - gvec[j];

    // vDot = Minv @ rhs
#pragma unroll
    for (int i = 0; i < 6; ++i) {
        float s = 0.0f;
#pragma unroll
        for (int j = 0; j < 6; ++j) s = fmaf(sMi[i*6 + j], rhs[j], s);
        xd[7 + i] = s;
    }
}

// ---------------------------------------------------------------------------
// Main kernel: one thread per row; LDS-staged coalesced float4 IO.
// ---------------------------------------------------------------------------
__global__ void fossen_rk2(const float* __restrict__ x, const float* __restrict__ u,
                           const float* __restrict__ mass, const float* __restrict__ volume,
                           const float* __restrict__ cog, const float* __restrict__ cob,
                           const float* __restrict__ mTot, const float* __restrict__ linDamp,
                           const float* __restrict__ linDampFow, const float* __restrict__ quadDamp,
                           const float* __restrict__ minv, float* __restrict__ out, int K)
{
    __shared__ __align__(16) float sx[TPB * 13];
    __shared__ __align__(16) float su[TPB * 6];
    __shared__ float sM[36], slD[36], slDF[36], sMi[36], sqD6[6], sc[8];

    const int tid = threadIdx.x;
    if (tid < 36) {
        sM [tid] = mTot[tid];
        slD[tid] = linDamp[tid];
        slDF[tid] = linDampFow[tid];
        sMi[tid] = minv[tid];
    }
    if (tid >= 36 && tid < 42) sqD6[tid - 36] = quadDamp[(tid - 36) * 7];
    if (tid == 42) { sc[0] = mass[0] * GRAV; sc[1] = volume[0] * DENS * GRAV; }
    if (tid >= 43 && tid < 46) sc[2 + tid - 43] = cog[tid - 43];
    if (tid >= 46 && tid < 49) sc[5 + tid - 46] = cob[tid - 46];

    const long r0  = (long)blockIdx.x * TPB;
    const bool full = (r0 + TPB) <= (long)K;
    const float* xb = x + r0 * 13;
    const float* ub = u + r0 * 6;

    if (full) {
        __builtin_prefetch(xb + TPB * 13, 0, 1);   // gfx1250 global_prefetch_b8
        const float4* x4 = (const float4*)xb;
        for (int i = tid; i < TPB * 13 / 4; i += TPB) ((float4*)sx)[i] = x4[i];
        const float4* u4 = (const float4*)ub;
        for (int i = tid; i < TPB * 6 / 4; i += TPB) ((float4*)su)[i] = u4[i];
    } else {
        const int n = (int)((long)K - r0);
        for (int i = tid; i < n * 13; i += TPB) sx[i] = xb[i];
        for (int i = tid; i < n * 6;  i += TPB) su[i] = ub[i];
    }
    __syncthreads();

    const long row = r0 + tid;
    float o[13];
    if (row < (long)K) {
        float xl[13], ul[6];
#pragma unroll
        for (int i = 0; i < 13; ++i) xl[i] = sx[tid * 13 + i];
#pragma unroll
        for (int i = 0; i < 6; ++i)  ul[i] = su[tid * 6 + i];

        float k1[13], xt[13], k2[13];
        fossen_dot(xl, ul, sM, slD, slDF, sqD6, sMi, sc, k1);
#pragma unroll
        for (int i = 0; i < 13; ++i) xt[i] = fmaf(DT_C, k1[i], xl[i]);
        fossen_dot(xt, ul, sM, slD, slDF, sqD6, sMi, sc, k2);
#pragma unroll
        for (int i = 0; i < 13; ++i) o[i] = fmaf(0.5f * DT_C, k1[i] + k2[i], xl[i]);

        const float s = fmaf(o[3], o[3], fmaf(o[4], o[4], fmaf(o[5], o[5], o[6] * o[6])));
        const float inv = 1.0f / sqrtf(s);
        o[3] *= inv; o[4] *= inv; o[5] *= inv; o[6] *= inv;
    }
    __syncthreads();
    if (row < (long)K) {
#pragma unroll
        for (int i = 0; i < 13; ++i) sx[tid * 13 + i] = o[i];
    }
    __syncthreads();

    if (full) {
        float4* o4 = (float4*)(out + r0 * 13);
        for (int i = tid; i < TPB * 13 / 4; i += TPB) o4[i] = ((float4*)sx)[i];
    } else {
        const int n = (int)((long)K - r0);
        for (int i = tid; i < n * 13; i += TPB) out[r0 * 13 + i] = sx[i];
    }
}

// ---------------------------------------------------------------------------
extern "C" void kernel_launch(void* const* d_in, const int* in_sizes, int n_in,
                              void* d_out, int out_size, void* d_ws, size_t ws_size,
                              hipStream_t stream)
{
    const float* x          = (const float*)d_in[0];
    const float* u          = (const float*)d_in[1];
    const float* mass       = (const float*)d_in[2];
    const float* volume     = (const float*)d_in[3];
    const float* cog        = (const float*)d_in[4];
    const float* cob        = (const float*)d_in[5];
    const float* mTot       = (const float*)d_in[6];
    const float* linDamp    = (const float*)d_in[7];
    const float* linDampFow = (const float*)d_in[8];
    const float* quadDamp   = (const float*)d_in[9];
    float* ws  = (float*)d_ws;
    float* out = (float*)d_out;

    const int K = in_sizes[0] / 13;

    prep_inverse<<<1, 32, 0, stream>>>(mTot, ws);

    const int blocks = (K + TPB - 1) / TPB;
    fossen_rk2<<<blocks, TPB, 0, stream>>>(x, u, mass, volume, cog, cob,
                                           mTot, linDamp, linDampFow, quadDamp,
                                           ws, out, K);
}